// Graph2GraphCNN_16965120819511
// MI455X (gfx1250) — compile-verified
//
#include <hip/hip_runtime.h>
#include <hip/hip_bf16.h>

typedef __attribute__((ext_vector_type(2))) float v2f;
typedef __attribute__((ext_vector_type(8))) float v8f;

#define NB 2
#define NN 360
#define CH 64
#define ROWS (NB * NN)            // 720
#define EDGES (NB * NN * NN)      // 259200
#define TILES (EDGES / 16)        // 16200 (exact)

static __device__ __forceinline__ v8f wmma_f32x4(v2f a, v2f b, v8f c) {
  // D = A(16x4 f32) * B(4x16 f32) + C(16x16 f32), wave32
  return __builtin_amdgcn_wmma_f32_16x16x4_f32(false, a, false, b, (short)0, c,
                                               false, false);
}

// Async global->LDS copy (4B per lane), tracked with ASYNCcnt.
// Low 32 bits of a generic pointer to __shared__ are the LDS byte offset.
static __device__ __forceinline__ void stage_to_lds_async(float* lds_dst,
                                                          const float* g_src) {
  unsigned lds_off = (unsigned)(uintptr_t)lds_dst;
  asm volatile("global_load_async_to_lds_b32 %0, %1, off"
               :
               : "v"(lds_off), "v"(g_src)
               : "memory");
}
static __device__ __forceinline__ void wait_async0() {
  asm volatile("s_wait_asynccnt 0x0" ::: "memory");
}

// ---------------- small node-side kernels (VALU, tiny) ----------------

__global__ void k_node0(const float* __restrict__ X, const float* __restrict__ W,
                        const float* __restrict__ bias, float* __restrict__ Y) {
  int t = blockIdx.x * blockDim.x + threadIdx.x;
  if (t >= ROWS * CH) return;
  int c = t & (CH - 1);
  int r = t >> 6;
  float s = bias[c];
#pragma unroll
  for (int k = 0; k < 4; ++k) s += X[r * 4 + k] * W[k * CH + c];
  Y[t] = s;
}

__global__ void k_agg(const float* __restrict__ xe, const float* __restrict__ xn,
                      float* __restrict__ agg) {
  int t = blockIdx.x * blockDim.x + threadIdx.x;
  if (t >= ROWS * CH) return;
  int c = t & (CH - 1);
  int r = t >> 6;          // r = b*NN + i
  int b = r / NN;
  const float* row = xe + (size_t)r * NN * CH + c;
  const float* xnb = xn + (size_t)b * NN * CH + c;
  float num = 0.f, den = 0.f;
  for (int j = 0; j < NN; ++j) {
    if (j + 8 < NN) __builtin_prefetch(row + (size_t)(j + 8) * CH, 0, 1);
    float e = row[(size_t)j * CH];
    num += e * xnb[(size_t)j * CH];
    den += e;
  }
  agg[t] = num / den;
}

__global__ void k_nodemid(const float* __restrict__ xn, const float* __restrict__ agg,
                          const float* __restrict__ W, const float* __restrict__ bias,
                          float* __restrict__ Y) {
  int t = blockIdx.x * blockDim.x + threadIdx.x;
  if (t >= ROWS * CH) return;
  int c = t & (CH - 1);
  int r = t >> 6;
  float s = bias[c];
  for (int k = 0; k < CH; ++k) s += xn[r * CH + k] * W[k * CH + c];
  for (int k = 0; k < CH; ++k) s += agg[r * CH + k] * W[(CH + k) * CH + c];
  Y[t] = s;
}

// Y[r,c] = sum_k X[r,k] * W[rowoff+k, c]   (X: 720x64, W: Kx64 row-major)
__global__ void k_proj(const float* __restrict__ X, const float* __restrict__ W,
                       int rowoff, float* __restrict__ Y) {
  int t = blockIdx.x * blockDim.x + threadIdx.x;
  if (t >= ROWS * CH) return;
  int c = t & (CH - 1);
  int r = t >> 6;
  float s = 0.f;
  for (int k = 0; k < CH; ++k) s += X[r * CH + k] * W[(rowoff + k) * CH + c];
  Y[t] = s;
}

__global__ void k_nodefinal(const float* __restrict__ xn, const float* __restrict__ agg,
                            const float* __restrict__ Wa, const float* __restrict__ ba,
                            const float* __restrict__ wb, const float* __restrict__ bb,
                            float* __restrict__ out) {
  __shared__ float red[CH];
  int r = blockIdx.x;
  int c = threadIdx.x;
  float s = ba[c];
  for (int k = 0; k < CH; ++k) s += xn[r * CH + k] * Wa[k * CH + c];
  for (int k = 0; k < CH; ++k) s += agg[r * CH + k] * Wa[(CH + k) * CH + c];
  s = fmaxf(s, 0.f) * wb[c];
  red[c] = s;
  __syncthreads();
#pragma unroll
  for (int off = 32; off > 0; off >>= 1) {
    if (c < off) red[c] += red[c + off];
    __syncthreads();
  }
  if (c == 0) out[r] = red[0] + bb[0];
}

// ---------------- WMMA edge kernels ----------------

// xe0 = edge_in(E,4) @ W(4,64) + bias : single 16x16x4 WMMA per (tile, n-tile)
#define WV2 8
__global__ void k_xe0(const float* __restrict__ E, const float* __restrict__ W,
                      const float* __restrict__ bias, float* __restrict__ out) {
  const int lane = threadIdx.x & 31;
  const int wave = threadIdx.x >> 5;
  const int e0 = (blockIdx.x * WV2 + wave) * 16;
  const int half = lane >> 4;
  const int lm = lane & 15;
  const float* arow = E + (size_t)(e0 + lm) * 4;
  v2f a;
  a.x = arow[2 * half];
  a.y = arow[2 * half + 1];
#pragma unroll
  for (int nt = 0; nt < 4; ++nt) {
    int n = nt * 16 + lm;
    v2f bw;
    bw.x = W[(2 * half + 0) * CH + n];
    bw.y = W[(2 * half + 1) * CH + n];
    float bv = bias[n];
    v8f c;
#pragma unroll
    for (int v = 0; v < 8; ++v) c[v] = bv;
    c = wmma_f32x4(a, bw, c);
    float* orow = out + (size_t)e0 * CH + n;
#pragma unroll
    for (int v = 0; v < 8; ++v) orow[(size_t)(v + 8 * half) * CH] = c[v];
  }
}

// xe1[e,:] = Pt[b,j,:] + Pm[b,i,:] + xe0[e,:] @ Wb(64,64) + bias   (K=64 WMMA)
#define WV5 8
__global__ void k_xemid(const float* __restrict__ xe0, const float* __restrict__ Pt,
                        const float* __restrict__ Pm, const float* __restrict__ Wb,
                        const float* __restrict__ bias, float* __restrict__ xe1) {
  __shared__ float Ws[CH * CH];
  for (int t = threadIdx.x; t < CH * CH; t += blockDim.x)
    stage_to_lds_async(&Ws[t], &Wb[t]);
  wait_async0();
  __syncthreads();
  const int lane = threadIdx.x & 31;
  const int wave = threadIdx.x >> 5;
  const int e0 = (blockIdx.x * WV5 + wave) * 16;
  const int half = lane >> 4;
  const int lm = lane & 15;
  const int b = e0 / (NN * NN);
  int ii[8], jj[8];
#pragma unroll
  for (int v = 0; v < 8; ++v) {
    int rem = (e0 + v + 8 * half) % (NN * NN);
    ii[v] = rem / NN;
    jj[v] = rem % NN;
  }
  const float* arow = xe0 + (size_t)(e0 + lm) * CH;
  v2f a[16];
#pragma unroll
  for (int s = 0; s < 16; ++s) {
    int k = 4 * s + 2 * half;
    a[s].x = arow[k];
    a[s].y = arow[k + 1];
  }
#pragma unroll
  for (int nt = 0; nt < 4; ++nt) {
    int n = nt * 16 + lm;
    v8f c;
#pragma unroll
    for (int v = 0; v < 8; ++v)
      c[v] = bias[n] + Pm[((size_t)b * NN + ii[v]) * CH + n] +
             Pt[((size_t)b * NN + jj[v]) * CH + n];
#pragma unroll
    for (int s = 0; s < 16; ++s) {
      int k = 4 * s + 2 * half;
      v2f bw;
      bw.x = Ws[k * CH + n];
      bw.y = Ws[(k + 1) * CH + n];
      c = wmma_f32x4(a[s], bw, c);
    }
    float* orow = xe1 + (size_t)e0 * CH + n;
#pragma unroll
    for (int v = 0; v < 8; ++v) orow[(size_t)(v + 8 * half) * CH] = c[v];
  }
}

// out[e,0:360] = relu(Qt[b,j]+Qm[b,i]+xe1[e]@Wbot+ba) @ W2(64,360) + b2, diag masked
#define WV8 4
__global__ void k_edgefinal(const float* __restrict__ xe1, const float* __restrict__ Qt,
                            const float* __restrict__ Qm, const float* __restrict__ Wbot,
                            const float* __restrict__ ba, const float* __restrict__ W2,
                            const float* __restrict__ b2, float* __restrict__ out) {
  __shared__ float Ws[CH * CH];
  __shared__ float heS[WV8][16 * CH];
  for (int t = threadIdx.x; t < CH * CH; t += blockDim.x)
    stage_to_lds_async(&Ws[t], &Wbot[t]);
  wait_async0();
  __syncthreads();
  const int lane = threadIdx.x & 31;
  const int wave = threadIdx.x >> 5;
  const int e0 = (blockIdx.x * WV8 + wave) * 16;
  const int half = lane >> 4;
  const int lm = lane & 15;
  const int b = e0 / (NN * NN);
  int ii[8], jj[8];
#pragma unroll
  for (int v = 0; v < 8; ++v) {
    int rem = (e0 + v + 8 * half) % (NN * NN);
    ii[v] = rem / NN;
    jj[v] = rem % NN;
  }
  const float* arow = xe1 + (size_t)(e0 + lm) * CH;
  v2f a[16];
#pragma unroll
  for (int s = 0; s < 16; ++s) {
    int k = 4 * s + 2 * half;
    a[s].x = arow[k];
    a[s].y = arow[k + 1];
  }
  float* hs = heS[wave];
  // GEMM 1: hidden(16x64) = relu(C + A @ Wbot)
#pragma unroll
  for (int nt = 0; nt < 4; ++nt) {
    int n = nt * 16 + lm;
    v8f c;
#pragma unroll
    for (int v = 0; v < 8; ++v)
      c[v] = ba[n] + Qm[((size_t)b * NN + ii[v]) * CH + n] +
             Qt[((size_t)b * NN + jj[v]) * CH + n];
#pragma unroll
    for (int s = 0; s < 16; ++s) {
      int k = 4 * s + 2 * half;
      v2f bw;
      bw.x = Ws[k * CH + n];
      bw.y = Ws[(k + 1) * CH + n];
      c = wmma_f32x4(a[s], bw, c);
    }
#pragma unroll
    for (int v = 0; v < 8; ++v)
      hs[(v + 8 * half) * CH + n] = fmaxf(c[v], 0.f);
  }
  // D-layout -> A-layout through per-wave LDS (same wave: LDS is in-order)
  v2f a2[16];
#pragma unroll
  for (int s = 0; s < 16; ++s) {
    int k = 4 * s + 2 * half;
    a2[s].x = hs[lm * CH + k];
    a2[s].y = hs[lm * CH + k + 1];
  }
  // GEMM 2: out(16x360) = hidden @ W2 + b2, 23 n-tiles (last half-masked)
  for (int nt = 0; nt < 23; ++nt) {
    int n = nt * 16 + lm;
    int nc = (n < NN) ? n : 0;  // clamp so loads stay in-bounds, EXEC stays full
    float bv = (n < NN) ? b2[nc] : 0.f;
    v8f c;
#pragma unroll
    for (int v = 0; v < 8; ++v) c[v] = bv;
#pragma unroll
    for (int s = 0; s < 16; ++s) {
      int k = 4 * s + 2 * half;
      v2f bw;
      bw.x = W2[(size_t)k * NN + nc];
      bw.y = W2[(size_t)(k + 1) * NN + nc];
      c = wmma_f32x4(a2[s], bw, c);
    }
    if (n < NN) {
#pragma unroll
      for (int v = 0; v < 8; ++v) {
        size_t e = (size_t)(e0 + v + 8 * half);
        out[e * NN + n] = (ii[v] == jj[v]) ? 0.f : c[v];
      }
    }
  }
}

// ---------------- launch ----------------

extern "C" void kernel_launch(void* const* d_in, const int* in_sizes, int n_in,
                              void* d_out, int out_size, void* d_ws, size_t ws_size,
                              hipStream_t stream) {
  (void)in_sizes; (void)n_in; (void)out_size; (void)ws_size;
  const float* node_in = (const float*)d_in[0];
  const float* edge_in = (const float*)d_in[1];
  const float* w_n0 = (const float*)d_in[2];
  const float* b_n0 = (const float*)d_in[3];
  const float* w_e0 = (const float*)d_in[4];
  const float* b_e0 = (const float*)d_in[5];
  const float* w_n1 = (const float*)d_in[6];
  const float* b_n1 = (const float*)d_in[7];
  const float* w_e1 = (const float*)d_in[8];
  const float* b_e1 = (const float*)d_in[9];
  const float* w_n2a = (const float*)d_in[10];
  const float* b_n2a = (const float*)d_in[11];
  const float* w_n2b = (const float*)d_in[12];
  const float* b_n2b = (const float*)d_in[13];
  const float* w_e2a = (const float*)d_in[14];
  const float* b_e2a = (const float*)d_in[15];
  const float* w_e2b = (const float*)d_in[16];
  const float* b_e2b = (const float*)d_in[17];

  float* ws = (float*)d_ws;
  float* xn0 = ws;
  float* agg0 = xn0 + ROWS * CH;
  float* xn1 = agg0 + ROWS * CH;
  float* agg1 = xn1 + ROWS * CH;
  float* Pt = agg1 + ROWS * CH;
  float* Pm = Pt + ROWS * CH;
  float* Qt = Pm + ROWS * CH;
  float* Qm = Qt + ROWS * CH;
  float* xe0 = Qm + ROWS * CH;
  float* xe1 = xe0 + (size_t)EDGES * CH;

  float* out_n = (float*)d_out;       // (2,360,1)
  float* out_e = out_n + ROWS;        // (2,360,360,360)

  const int vec_blocks = (ROWS * CH + 255) / 256;  // 180

  k_node0<<<vec_blocks, 256, 0, stream>>>(node_in, w_n0, b_n0, xn0);
  k_xe0<<<TILES / WV2, 256, 0, stream>>>(edge_in, w_e0, b_e0, xe0);
  k_agg<<<vec_blocks, 256, 0, stream>>>(xe0, xn0, agg0);
  k_nodemid<<<vec_blocks, 256, 0, stream>>>(xn0, agg0, w_n1, b_n1, xn1);
  k_proj<<<vec_blocks, 256, 0, stream>>>(xn0, w_e1, 0, Pt);
  k_proj<<<vec_blocks, 256, 0, stream>>>(xn0, w_e1, CH, Pm);
  k_xemid<<<TILES / WV5, 256, 0, stream>>>(xe0, Pt, Pm, w_e1 + 2 * CH * CH, b_e1, xe1);
  k_agg<<<vec_blocks, 256, 0, stream>>>(xe1, xn1, agg1);
  k_proj<<<vec_blocks, 256, 0, stream>>>(xn1, w_e2a, 0, Qt);
  k_proj<<<vec_blocks, 256, 0, stream>>>(xn1, w_e2a, CH, Qm);
  k_nodefinal<<<ROWS, CH, 0, stream>>>(xn1, agg1, w_n2a, b_n2a, w_n2b, b_n2b, out_n);
  k_edgefinal<<<TILES / WV8, 128, 0, stream>>>(xe1, Qt, Qm, w_e2a + 2 * CH * CH,
                                               b_e2a, w_e2b, b_e2b, out_e);
}